// GCN_subatt_test_86887188398718
// MI455X (gfx1250) — compile-verified
//
#include <hip/hip_runtime.h>
#include <math.h>

#define NN      100000
#define EE      1600000
#define NFEATC  256
#define NHIDC   64
#define NCLASSC 40
#define NSTRUCC 16

typedef __attribute__((ext_vector_type(2))) float v2f;
typedef __attribute__((ext_vector_type(8))) float v8f;

__device__ __forceinline__ int f2oi(float f) {
  int i = __float_as_int(f);
  return (i >= 0) ? i : (i ^ 0x7FFFFFFF);
}
__device__ __forceinline__ float oi2f(int i) {
  int j = (i >= 0) ? i : (i ^ 0x7FFFFFFF);
  return __int_as_float(j);
}

// ---------------------------------------------------------------------------
// 0) zero the scatter accumulators + init global-softmax scalars
// ---------------------------------------------------------------------------
__global__ __launch_bounds__(256) void gcn_fill(float* __restrict__ agg1,
                                                float* __restrict__ agg2,
                                                int* __restrict__ dmax,
                                                float* __restrict__ dsum) {
  int t = blockIdx.x * 256 + threadIdx.x;
  if (t < NN * NHIDC)   agg1[t] = 0.0f;
  if (t < NN * NCLASSC) agg2[t] = 0.0f;
  if (t == 0) { *dmax = f2oi(-INFINITY); *dsum = 0.0f; }
}

// ---------------------------------------------------------------------------
// 1) support1 = x @ W1   via V_WMMA_F32_16X16X4_F32  (exact f32 path)
//    block = 128 threads = 4 waves; each wave owns a 16x64 output tile.
//    W1 staged in LDS pair-interleaved: lwp[k/2][col] = {W1[k][col], W1[k+1][col]}
//    so each B fragment is a single ds_load_b64 straight into the even-aligned
//    VGPR pair the WMMA consumes (no repack moves).
// ---------------------------------------------------------------------------
__global__ __launch_bounds__(128) void gcn_gemm1(const float* __restrict__ x,
                                                 const float* __restrict__ W1,
                                                 float* __restrict__ sup) {
  __shared__ v2f lwp[(NFEATC / 2) * NHIDC];  // 128 pair-rows x 64 cols = 64KB
  for (int i = threadIdx.x; i < (NFEATC / 2) * NHIDC; i += 128) {
    int p = i >> 6;   // pair row (k = 2p, 2p+1)
    int c = i & 63;   // column
    v2f t;
    t.x = W1[(size_t)(2 * p)     * NHIDC + c];
    t.y = W1[(size_t)(2 * p + 1) * NHIDC + c];
    lwp[i] = t;
  }
  __syncthreads();

  const int wave = threadIdx.x >> 5;
  const int lane = threadIdx.x & 31;
  const int half = lane >> 4;   // which 16-lane half of the wave
  const int lr   = lane & 15;
  const int tileRow = blockIdx.x * 64 + wave * 16;

  // A fragment rows: clamp so out-of-range rows read valid memory (results discarded)
  const int arow = min(tileRow + lr, NN - 1);
  const float2* xr = (const float2*)(x + (size_t)arow * NFEATC);

  v8f acc0 = {}, acc1 = {}, acc2 = {}, acc3 = {};
  for (int k = 0; k < NFEATC; k += 4) {
    // A 16x4 f32 layout: lanes 0-15 -> K = k,k+1 ; lanes 16-31 -> K = k+2,k+3
    float2 av = xr[(k >> 1) + half];
    v2f a; a.x = av.x; a.y = av.y;

    const v2f* bp = lwp + ((k >> 1) + half) * NHIDC + lr;
    v2f b0 = bp[0];
    v2f b1 = bp[16];
    v2f b2 = bp[32];
    v2f b3 = bp[48];

    acc0 = __builtin_amdgcn_wmma_f32_16x16x4_f32(false, a, false, b0, (short)0, acc0, false, false);
    acc1 = __builtin_amdgcn_wmma_f32_16x16x4_f32(false, a, false, b1, (short)0, acc1, false, false);
    acc2 = __builtin_amdgcn_wmma_f32_16x16x4_f32(false, a, false, b2, (short)0, acc2, false, false);
    acc3 = __builtin_amdgcn_wmma_f32_16x16x4_f32(false, a, false, b3, (short)0, acc3, false, false);
  }

  // C/D layout: VGPR i -> M = i (lanes 0-15) / M = i+8 (lanes 16-31), N = lr
#pragma unroll
  for (int i = 0; i < 8; ++i) {
    int row = tileRow + half * 8 + i;
    if (row < NN) {
      float* o = sup + (size_t)row * NHIDC + lr;
      o[0]  = acc0[i];
      o[16] = acc1[i];
      o[32] = acc2[i];
      o[48] = acc3[i];
    }
  }
}

// ---------------------------------------------------------------------------
// 2) scatter-add layer1: agg1[dst] += support1[src] * w   (64 feats/edge)
// ---------------------------------------------------------------------------
__global__ __launch_bounds__(256) void gcn_scatter1(const float* __restrict__ sup,
                                                    const int* __restrict__ ei,
                                                    const float* __restrict__ ew,
                                                    float* __restrict__ agg) {
  int t = blockIdx.x * 256 + threadIdx.x;
  if (t >= EE * NHIDC) return;
  int e = t >> 6;
  int f = t & 63;
  int s = ei[EE + e];  // row 1 = src
  int d = ei[e];       // row 0 = dst
  atomicAdd(&agg[(size_t)d * NHIDC + f], sup[(size_t)s * NHIDC + f] * ew[e]);
}

// ---------------------------------------------------------------------------
// 3) h = relu(agg1+b1); support2 = h@W2; y = h@enc_W+enc_b -> out;
//    alog = h@att -> ws; global atomicMax of alog
// ---------------------------------------------------------------------------
__global__ __launch_bounds__(256) void gcn_post1(const float* __restrict__ agg1,
                                                 const float* __restrict__ b1,
                                                 const float* __restrict__ W2,
                                                 const float* __restrict__ encW,
                                                 const float* __restrict__ encb,
                                                 const float* __restrict__ att,
                                                 float* __restrict__ sup2,
                                                 float* __restrict__ yout,
                                                 float* __restrict__ alog,
                                                 int* __restrict__ dmax) {
  __shared__ float lW2[NHIDC * NCLASSC];
  __shared__ float lE[NHIDC * NSTRUCC];
  __shared__ float lA[NHIDC * NSTRUCC];
  __shared__ float lb1[NHIDC];
  __shared__ float leb[NSTRUCC];
  for (int i = threadIdx.x; i < NHIDC * NCLASSC; i += 256) lW2[i] = W2[i];
  for (int i = threadIdx.x; i < NHIDC * NSTRUCC; i += 256) lE[i] = encW[i];
  for (int i = threadIdx.x; i < NHIDC * NSTRUCC; i += 256) lA[i] = att[i];
  if (threadIdx.x < NHIDC)   lb1[threadIdx.x] = b1[threadIdx.x];
  if (threadIdx.x < NSTRUCC) leb[threadIdx.x] = encb[threadIdx.x];
  __syncthreads();

  int n = blockIdx.x * 256 + threadIdx.x;
  if (n >= NN) return;

  float h[NHIDC];
#pragma unroll
  for (int k = 0; k < NHIDC; ++k) {
    float v = agg1[(size_t)n * NHIDC + k] + lb1[k];
    h[k] = v > 0.0f ? v : 0.0f;
  }

  for (int j = 0; j < NCLASSC; ++j) {
    float acc = 0.0f;
#pragma unroll
    for (int k = 0; k < NHIDC; ++k) acc = fmaf(h[k], lW2[k * NCLASSC + j], acc);
    sup2[(size_t)n * NCLASSC + j] = acc;
  }

  for (int j = 0; j < NSTRUCC; ++j) {
    float acc = leb[j];
#pragma unroll
    for (int k = 0; k < NHIDC; ++k) acc = fmaf(h[k], lE[k * NSTRUCC + j], acc);
    yout[(size_t)n * NSTRUCC + j] = acc;
  }

  float lm = -INFINITY;
  for (int j = 0; j < NSTRUCC; ++j) {
    float acc = 0.0f;
#pragma unroll
    for (int k = 0; k < NHIDC; ++k) acc = fmaf(h[k], lA[k * NSTRUCC + j], acc);
    alog[(size_t)n * NSTRUCC + j] = acc;
    lm = fmaxf(lm, acc);
  }
  atomicMax(dmax, f2oi(lm));
}

// ---------------------------------------------------------------------------
// 4) scatter-add layer2: agg2[dst] += support2[src] * w   (40 feats/edge)
// ---------------------------------------------------------------------------
__global__ __launch_bounds__(256) void gcn_scatter2(const float* __restrict__ sup,
                                                    const int* __restrict__ ei,
                                                    const float* __restrict__ ew,
                                                    float* __restrict__ agg) {
  int t = blockIdx.x * 256 + threadIdx.x;
  if (t >= EE * NCLASSC) return;
  int e = t / NCLASSC;
  int f = t - e * NCLASSC;
  int s = ei[EE + e];
  int d = ei[e];
  atomicAdd(&agg[(size_t)d * NCLASSC + f], sup[(size_t)s * NCLASSC + f] * ew[e]);
}

// ---------------------------------------------------------------------------
// 5) logits = agg2 + b2 ; log_softmax per row -> out[0 : N*40]
// ---------------------------------------------------------------------------
__global__ __launch_bounds__(256) void gcn_post2(const float* __restrict__ agg2,
                                                 const float* __restrict__ b2,
                                                 float* __restrict__ out) {
  int n = blockIdx.x * 256 + threadIdx.x;
  if (n >= NN) return;
  float v[NCLASSC];
  float m = -INFINITY;
#pragma unroll
  for (int j = 0; j < NCLASSC; ++j) {
    v[j] = agg2[(size_t)n * NCLASSC + j] + b2[j];
    m = fmaxf(m, v[j]);
  }
  float s = 0.0f;
#pragma unroll
  for (int j = 0; j < NCLASSC; ++j) s += expf(v[j] - m);
  float ls = logf(s);
#pragma unroll
  for (int j = 0; j < NCLASSC; ++j) out[(size_t)n * NCLASSC + j] = v[j] - m - ls;
}

// ---------------------------------------------------------------------------
// 6) global softmax over flattened alog: sum of exp(v - max)
// ---------------------------------------------------------------------------
__global__ __launch_bounds__(256) void gcn_sumexp(const float* __restrict__ alog,
                                                  const int* __restrict__ dmax,
                                                  float* __restrict__ dsum) {
  int t = blockIdx.x * 256 + threadIdx.x;
  float m = oi2f(*dmax);
  float v = 0.0f;
  if (t < NN * NSTRUCC) v = expf(alog[t] - m);
#pragma unroll
  for (int off = 16; off > 0; off >>= 1) v += __shfl_down(v, off, 32);
  if ((threadIdx.x & 31) == 0) atomicAdd(dsum, v);
}

// ---------------------------------------------------------------------------
// 7) al = exp(alog - max) / sum -> out[N*56 : N*72]
// ---------------------------------------------------------------------------
__global__ __launch_bounds__(256) void gcn_alout(const float* __restrict__ alog,
                                                 const int* __restrict__ dmax,
                                                 const float* __restrict__ dsum,
                                                 float* __restrict__ out_al) {
  int t = blockIdx.x * 256 + threadIdx.x;
  if (t >= NN * NSTRUCC) return;
  float m = oi2f(*dmax);
  float s = *dsum;
  out_al[t] = expf(alog[t] - m) / s;
}

// ---------------------------------------------------------------------------
extern "C" void kernel_launch(void* const* d_in, const int* in_sizes, int n_in,
                              void* d_out, int out_size, void* d_ws, size_t ws_size,
                              hipStream_t stream) {
  (void)in_sizes; (void)n_in; (void)out_size; (void)ws_size;

  const float* x    = (const float*)d_in[0];
  const int*   ei   = (const int*)  d_in[1];
  const float* ew   = (const float*)d_in[2];
  const float* W1   = (const float*)d_in[3];
  const float* b1   = (const float*)d_in[4];
  const float* W2   = (const float*)d_in[5];
  const float* b2   = (const float*)d_in[6];
  const float* encW = (const float*)d_in[7];
  const float* encb = (const float*)d_in[8];
  const float* att  = (const float*)d_in[9];

  float* out = (float*)d_out;
  float* out_logits = out;                                    // [N,40]
  float* out_y      = out + (size_t)NN * NCLASSC;             // [N,16]
  float* out_al     = out + (size_t)NN * (NCLASSC + NSTRUCC); // [N,16]

  float* ws   = (float*)d_ws;
  float* sup1 = ws;                                     // N*64
  float* agg1 = sup1 + (size_t)NN * NHIDC;              // N*64
  float* sup2 = agg1 + (size_t)NN * NHIDC;              // N*40
  float* agg2 = sup2 + (size_t)NN * NCLASSC;            // N*40
  float* alog = agg2 + (size_t)NN * NCLASSC;            // N*16
  int*   dmax = (int*)  (alog + (size_t)NN * NSTRUCC);
  float* dsum = (float*)(dmax + 1);

  int g_fill  = (NN * NHIDC + 255) / 256;
  int g_gemm  = (NN + 63) / 64;
  int g_sc1   = (int)(((long long)EE * NHIDC   + 255) / 256);
  int g_sc2   = (int)(((long long)EE * NCLASSC + 255) / 256);
  int g_rows  = (NN + 255) / 256;
  int g_al    = (NN * NSTRUCC + 255) / 256;

  gcn_fill    <<<g_fill, 256, 0, stream>>>(agg1, agg2, dmax, dsum);
  gcn_gemm1   <<<g_gemm, 128, 0, stream>>>(x, W1, sup1);
  gcn_scatter1<<<g_sc1,  256, 0, stream>>>(sup1, ei, ew, agg1);
  gcn_post1   <<<g_rows, 256, 0, stream>>>(agg1, b1, W2, encW, encb, att,
                                           sup2, out_y, alog, dmax);
  gcn_scatter2<<<g_sc2,  256, 0, stream>>>(sup2, ei, ew, agg2);
  gcn_post2   <<<g_rows, 256, 0, stream>>>(agg2, b2, out_logits);
  gcn_sumexp  <<<g_al,   256, 0, stream>>>(alog, dmax, dsum);
  gcn_alout   <<<g_al,   256, 0, stream>>>(alog, dmax, dsum, out_al);
}